// Legendre2_21371757265202
// MI455X (gfx1250) — compile-verified
//
#include <hip/hip_runtime.h>

typedef __attribute__((ext_vector_type(2))) float v2f;
typedef __attribute__((ext_vector_type(8))) float v8f;

// Problem constants (reference: N=524288, D=64, K=64, O=32, DEGREE=6)
#define NROWS 524288
#define DDIM  64
#define KDIM  64
#define ODIM  32

// Workspace layout (floats):
//   Bswz: 6 gemms * 4 ntiles * 16 ktiles * 32 lanes * 2 = 24576 floats (96 KB)
//   Cswz: 2 ntiles * 16 ktiles * 32 lanes * 2          =  2048 floats ( 8 KB)
#define CSWZ_OFF 24576

// ---------------------------------------------------------------------------
// Prep: fold row-normalization + Legendre coefficient into B = coef * Ti^T,
// and emit B (and C^T) pre-swizzled into WMMA 16x16x4 B-fragment order:
//   frag(g, nt, kt, lane) = { B[d][n], B[d+1][n] }
//   with d = kt*4 + (lane<16?0:2), n = nt*16 + (lane&15)
// ---------------------------------------------------------------------------
__global__ void legendre_prep(const float* __restrict__ T,
                              const float* __restrict__ C,
                              float* __restrict__ ws) {
    const int g = blockIdx.x;   // 0..5 = degree matrices, 6 = C matrix
    const int t = threadIdx.x;  // 64 threads
    if (g < 6) {
        const int k = t;                     // output column of this gemm
        const float* Tg = T + g * (KDIM * DDIM);
        float scale = 1.0f;
        if (g >= 1) {                        // i = g+1 >= 2: row-normalize + coef
            float s = 0.0f;
            for (int d = 0; d < DDIM; ++d) s += Tg[k * DDIM + d];
            const float fi = (float)(g + 1);
            scale = ((2.0f * fi - 1.0f) / fi) / s;
        }
        const int nt = k >> 4;
        for (int d = 0; d < DDIM; ++d) {
            const float val  = scale * Tg[k * DDIM + d];
            const int   kt   = d >> 2;
            const int   lane = (k & 15) + ((d & 2) << 3);  // +16 if d%4 in {2,3}
            const int   sub  = d & 1;
            ws[((((g * 4 + nt) * 16 + kt) * 32 + lane) << 1) + sub] = val;
        }
    } else if (t < ODIM) {                   // C^T fragments for the final gemm
        const int o  = t;
        const int nt = o >> 4;
        for (int d = 0; d < DDIM; ++d) {
            const float val  = C[o * DDIM + d];   // Ct[d][o]
            const int   kt   = d >> 2;
            const int   lane = (o & 15) + ((d & 2) << 3);
            const int   sub  = d & 1;
            ws[CSWZ_OFF + (((nt * 16 + kt) * 32 + lane) << 1) + sub] = val;
        }
    }
}

// ---------------------------------------------------------------------------
// Main: 8 waves / block, each wave owns a 16-row tile of z.
// z fragments live in VGPRs for all 6 GEMMs; recurrence is elementwise on the
// WMMA C/D accumulator layout. Final p @ C^T transposes p via a per-wave LDS
// round-trip (D-layout store -> A-fragment load).
// ---------------------------------------------------------------------------
__global__ __launch_bounds__(256)
void legendre_main(const float* __restrict__ z,
                   const float* __restrict__ ws,
                   const float* __restrict__ beta,
                   float* __restrict__ out) {
    __shared__ float lds[8 * 16 * DDIM];  // 32 KB: 4 KB per wave

    const int wave = threadIdx.x >> 5;
    const int lane = threadIdx.x & 31;
    const int mrow = lane & 15;
    const int koff = (lane < 16) ? 0 : 2;
    const int row0 = blockIdx.x * 128 + wave * 16;

    // --- load z tile in A-fragment order: 16 x float2 per lane ---
    v2f zf[16];
    const float* zp = z + (size_t)(row0 + mrow) * DDIM + koff;
#pragma unroll
    for (int kt = 0; kt < 16; ++kt)
        zf[kt] = *(const v2f*)(zp + kt * 4);

    const v2f* Bswz = (const v2f*)ws;
    const v2f* Cswz = (const v2f*)(ws + CSWZ_OFF);

    v8f pprev[4], pcurr[4];
#pragma unroll
    for (int nt = 0; nt < 4; ++nt)
        pprev[nt] = (v8f){1.f, 1.f, 1.f, 1.f, 1.f, 1.f, 1.f, 1.f};  // P0 = 1

    // --- degree 1: P1 = z @ T1^T ---
#pragma unroll
    for (int nt = 0; nt < 4; ++nt) {
        v8f acc = (v8f){0.f, 0.f, 0.f, 0.f, 0.f, 0.f, 0.f, 0.f};
        const v2f* bp = Bswz + (size_t)(0 * 4 + nt) * 16 * 32 + lane;
#pragma unroll
        for (int kt = 0; kt < 16; ++kt) {
            acc = __builtin_amdgcn_wmma_f32_16x16x4_f32(
                false, zf[kt], false, bp[kt * 32], (short)0, acc, false, false);
        }
        pcurr[nt] = acc;
    }

    // --- degrees 2..6: P_i = (z @ coef*Ti'^T) * P_{i-1} - ((i-1)/i) * P_{i-2} ---
#pragma unroll
    for (int i = 2; i <= 6; ++i) {
        const int   g  = i - 1;
        const float c2 = (float)(i - 1) / (float)i;
#pragma unroll
        for (int nt = 0; nt < 4; ++nt) {
            v8f acc = (v8f){0.f, 0.f, 0.f, 0.f, 0.f, 0.f, 0.f, 0.f};
            const v2f* bp = Bswz + (size_t)(g * 4 + nt) * 16 * 32 + lane;
#pragma unroll
            for (int kt = 0; kt < 16; ++kt) {
                acc = __builtin_amdgcn_wmma_f32_16x16x4_f32(
                    false, zf[kt], false, bp[kt * 32], (short)0, acc, false, false);
            }
            const v8f pnew = acc * pcurr[nt] - pprev[nt] * c2;
            pprev[nt] = pcurr[nt];
            pcurr[nt] = pnew;
        }
    }

    // --- transpose P via LDS: D-layout scatter, A-fragment gather ---
    float* myl = lds + wave * 16 * DDIM;
    const int mbase = (lane < 16) ? 0 : 8;
#pragma unroll
    for (int nt = 0; nt < 4; ++nt) {
#pragma unroll
        for (int j = 0; j < 8; ++j)
            myl[(mbase + j) * DDIM + nt * 16 + mrow] = pcurr[nt][j];
    }
    __syncthreads();

    v2f pf[16];
    const float* pr = myl + mrow * DDIM + koff;
#pragma unroll
    for (int kt = 0; kt < 16; ++kt)
        pf[kt] = *(const v2f*)(pr + kt * 4);

    // --- out = P @ C^T + beta ---
#pragma unroll
    for (int nt = 0; nt < 2; ++nt) {
        v8f acc = (v8f){0.f, 0.f, 0.f, 0.f, 0.f, 0.f, 0.f, 0.f};
        const v2f* cp = Cswz + (size_t)nt * 16 * 32 + lane;
#pragma unroll
        for (int kt = 0; kt < 16; ++kt) {
            acc = __builtin_amdgcn_wmma_f32_16x16x4_f32(
                false, pf[kt], false, cp[kt * 32], (short)0, acc, false, false);
        }
        const float b = beta[nt * 16 + mrow];
        acc = acc + b;
        float* op = out + (size_t)(row0 + mbase) * ODIM + nt * 16 + mrow;
#pragma unroll
        for (int j = 0; j < 8; ++j)
            op[j * ODIM] = acc[j];
    }
}

extern "C" void kernel_launch(void* const* d_in, const int* in_sizes, int n_in,
                              void* d_out, int out_size, void* d_ws, size_t ws_size,
                              hipStream_t stream) {
    const float* z    = (const float*)d_in[0];
    const float* T    = (const float*)d_in[1];
    const float* C    = (const float*)d_in[2];
    const float* beta = (const float*)d_in[3];
    float*       ws   = (float*)d_ws;

    legendre_prep<<<7, 64, 0, stream>>>(T, C, ws);
    legendre_main<<<NROWS / 128, 256, 0, stream>>>(z, ws, beta, (float*)d_out);
}